// CayleyFilter_46222438039786
// MI455X (gfx1250) — compile-verified
//
#include <hip/hip_runtime.h>
#include <hip/hip_bf16.h>

typedef __attribute__((ext_vector_type(2))) float v2f;
typedef __attribute__((ext_vector_type(8))) float v8f;

// ---------------------------------------------------------------------------
// Problem constants (from reference): N=32, C=32, m=64, M=4096, ORDER=8, Cout=64
//   rows = N*M = 131072, Kr = 9*32 = 288 (per real/imag), cols = 64
// ---------------------------------------------------------------------------
#define NBATCH   32
#define CIN      32
#define MVERT    4096
#define NORD     9          // ORDER+1
#define COUT     64
#define KTOT     288        // NORD*CIN
#define KSTEPS   72         // KTOT/4  (fp32 WMMA K=4)
// Prepacked B-fragments: [col_tile(4)][kstep(72)][lane(32)] of float2, per matrix
#define BFRAG_F2 (4*KSTEPS*32)   // 9216 float2 per matrix

// ---------------------------------------------------------------------------
// Prepack 2*Wr / 2*Wi into per-lane V_WMMA_F32_16X16X4_F32 B-fragment order.
// B is K(4) x N(16): lanes 0-15 hold {K0,K1} for N=lane, lanes 16-31 hold
// {K2,K3} for N=lane-16.  Factor 2.0 from "out = 2*(...)" folded in here.
// ---------------------------------------------------------------------------
__global__ void cayley_prepack(const float* __restrict__ wr,
                               const float* __restrict__ wi,
                               v2f* __restrict__ bfr,
                               v2f* __restrict__ bfi) {
    int g = blockIdx.x * 256 + threadIdx.x;
    if (g >= 2 * BFRAG_F2) return;
    int mat  = g / BFRAG_F2;
    int r    = g % BFRAG_F2;
    int ct   = r / (KSTEPS * 32);
    int r2   = r % (KSTEPS * 32);
    int s    = r2 / 32;
    int lane = r2 % 32;
    int half = lane >> 4;
    int c    = ct * 16 + (lane & 15);
    int kk   = 4 * s + 2 * half;
    const float* w = mat ? wi : wr;
    v2f val;
    val.x = 2.0f * w[(size_t)kk * COUT + c];
    val.y = 2.0f * w[(size_t)(kk + 1) * COUT + c];
    (mat ? bfi : bfr)[r] = val;
}

// ---------------------------------------------------------------------------
// Fused Cayley-feature x WMMA-GEMM kernel.
// Block = 128 threads (4 waves). Block owns 64 consecutive global rows
// (one batch n, vertices v0..v0+63).  Wave w computes cols [16w,16w+16) for
// all four 16-row subtiles, chaining real & imag fp32 WMMAs into one acc.
// ---------------------------------------------------------------------------
__global__ __launch_bounds__(128) void cayley_main(
        const float* __restrict__ x,      // (N, CIN, MVERT)
        const float* __restrict__ diagL,  // (MVERT)
        const v2f*  __restrict__ bfr,     // prepacked real B frags
        const v2f*  __restrict__ bfi,     // prepacked imag B frags
        float* __restrict__ out)          // (N*MVERT, COUT)
{
    __shared__ float xs[CIN * 64];       // x[n][ci][v0..v0+63]
    __shared__ float prs[NORD * 64];     // Re(c_v^k)
    __shared__ float pis[NORD * 64];     // Im(c_v^k)

    const int tid = threadIdx.x;
    const int r0  = blockIdx.x * 64;     // first global row
    const int n   = r0 >> 12;            // / MVERT
    const int v0  = r0 & (MVERT - 1);

    // ---- stage x tile (32 ci x 64 v = 8KB) with float4 loads ----
    for (int i = tid; i < CIN * 16; i += 128) {
        int ci = i >> 4;                 // 16 float4 per ci-row
        int fi = i & 15;
        const float4 val = *(const float4*)(x + (((size_t)n * CIN + ci) << 12)
                                              + v0 + (fi << 2));
        ((float4*)xs)[i] = val;
    }
    // ---- per-vertex Cayley powers c^k, c=(d-i)/(d+i)=((d^2-1) - 2di)/(d^2+1) ----
    if (tid < 64) {
        float d   = diagL[v0 + tid];
        float den = d * d + 1.0f;
        float cr  = (d * d - 1.0f) / den;
        float cm  = -2.0f * d / den;
        float pr = 1.0f, pi = 0.0f;
        #pragma unroll
        for (int k = 0; k < NORD; ++k) {
            prs[k * 64 + tid] = pr;
            pis[k * 64 + tid] = pi;
            float nr = pr * cr - pi * cm;
            float ni = pr * cm + pi * cr;
            pr = nr; pi = ni;
        }
    }
    __syncthreads();

    const int lane = tid & 31;
    const int wv   = tid >> 5;           // col tile 0..3
    const int half = lane >> 4;          // A/B fragment half (K01 vs K23)
    const int row  = lane & 15;          // M within subtile / N within col tile

    v8f acc[4];
    #pragma unroll
    for (int s = 0; s < 4; ++s) {
        v8f z = {0.f,0.f,0.f,0.f,0.f,0.f,0.f,0.f};
        acc[s] = z;
    }

    const v2f* brp = bfr + (size_t)wv * KSTEPS * 32 + lane;
    const v2f* bip = bfi + (size_t)wv * KSTEPS * 32 + lane;

    for (int k = 0; k < NORD; ++k) {
        float prk[4], pik[4];
        #pragma unroll
        for (int s = 0; s < 4; ++s) {
            prk[s] = prs[k * 64 + s * 16 + row];
            pik[s] = pis[k * 64 + s * 16 + row];
        }
        #pragma unroll
        for (int t = 0; t < 8; ++t) {          // 8 K-steps of 4 per order
            const int ci0 = t * 4 + 2 * half;  // this lane-half's first ci
            v2f br = brp[(k * 8 + t) * 32];
            v2f bi = bip[(k * 8 + t) * 32];
            #pragma unroll
            for (int s = 0; s < 4; ++s) {      // 4 row subtiles share B frag
                const int vL = s * 16 + row;
                float x0 = xs[ci0 * 64 + vL];
                float x1 = xs[ci0 * 64 + 64 + vL];
                v2f ar; ar.x = prk[s] * x0; ar.y = prk[s] * x1;
                v2f ai; ai.x = pik[s] * x0; ai.y = pik[s] * x1;
                acc[s] = __builtin_amdgcn_wmma_f32_16x16x4_f32(
                            false, ar, false, br, (short)0, acc[s], false, false);
                acc[s] = __builtin_amdgcn_wmma_f32_16x16x4_f32(
                            false, ai, false, bi, (short)0, acc[s], false, false);
            }
        }
    }

    // ---- store 16x16 f32 D tiles: VGPR j -> M=j (+8 for lanes 16-31) ----
    const int c = wv * 16 + row;
    #pragma unroll
    for (int s = 0; s < 4; ++s) {
        #pragma unroll
        for (int j = 0; j < 8; ++j) {
            int rr = r0 + s * 16 + half * 8 + j;
            out[(size_t)rr * COUT + c] = acc[s][j];
        }
    }
}

extern "C" void kernel_launch(void* const* d_in, const int* in_sizes, int n_in,
                              void* d_out, int out_size, void* d_ws, size_t ws_size,
                              hipStream_t stream) {
    const float* x     = (const float*)d_in[0];
    const float* wr    = (const float*)d_in[1];
    const float* wi    = (const float*)d_in[2];
    const float* diagL = (const float*)d_in[3];
    // d_in[4..6] (vals, rows, cols) unused: the off-diagonal SpMM cancels
    // exactly in the reference recurrence (tr = (offr + d*zr) + zi - offr).
    float* out = (float*)d_out;

    v2f* bfr = (v2f*)d_ws;               // 9216 float2 = 72KB
    v2f* bfi = bfr + BFRAG_F2;           // next 72KB   (ws usage: 144KB)

    cayley_prepack<<<(2 * BFRAG_F2 + 255) / 256, 256, 0, stream>>>(wr, wi, bfr, bfi);
    cayley_main<<<(NBATCH * MVERT) / 64, 128, 0, stream>>>(x, diagL, bfr, bfi, out);
}